// MoELoRALinear_68813966016590
// MI455X (gfx1250) — compile-verified
//
#include <hip/hip_runtime.h>

// Problem constants (from reference): B=4, S=4096, D=1024, E=10, R=4, TOP_K=2
#define NTOK    16384          // B*S
#define DDIM    1024
#define NEXP    10
#define RANK    4
#define KLORA   40             // E*R
#define WGROWS  64             // padded rows: [0..9]=gate_W, [10..49]=lora_down, rest zero
#define TM      128
#define TN      128
#define KT      32
#define NCHUNK  (DDIM / KT)    // 32
#define SCALING 0.25f          // 1/R

typedef float v2f __attribute__((ext_vector_type(2)));
typedef float v8f __attribute__((ext_vector_type(8)));

// V_WMMA_F32_16X16X4_F32: D(16x16 f32, 8 VGPR) = A(16x4 f32) x B(4x16 f32) + C
static __device__ __forceinline__ v8f wmma_fp32(v2f a, v2f b, v8f c) {
    return __builtin_amdgcn_wmma_f32_16x16x4_f32(false, a, false, b, (short)0, c, false, false);
}

// CDNA5 async global->LDS copy (ASYNCcnt-tracked, no VGPR round trip).
// GVS addressing: saddr = uniform 64-bit base (SGPR pair), vaddr = 32-bit byte offset.
// First operand = LDS byte address VGPR (VDST).
static __device__ __forceinline__ void async_b128_to_lds(unsigned lds_addr,
                                                         const float* gbase,
                                                         unsigned byte_off) {
    asm volatile("global_load_async_to_lds_b128 %0, %1, %2 offset:0"
                 :: "v"(lds_addr), "v"(byte_off),
                    "s"((unsigned long long)(uintptr_t)gbase)
                 : "memory");
}

#define WAIT_ASYNCCNT(N) asm volatile("s_wait_asynccnt %0" :: "i"(N) : "memory")

// ---------------------------------------------------------------------------
// Kernel 1: pack Wg[64x1024] (gate + lora_down, zero-padded) and UpT[1024x40]
// (UpT[o][e*4+r] = lora_up[e][o][r])
// ---------------------------------------------------------------------------
__global__ void pack_kernel(const float* __restrict__ gate_W,
                            const float* __restrict__ lora_down,
                            const float* __restrict__ lora_up,
                            float* __restrict__ Wg,
                            float* __restrict__ UpT) {
    int idx = blockIdx.x * 256 + threadIdx.x;
    if (idx < WGROWS * DDIM) {
        int row = idx / DDIM;
        int col = idx - row * DDIM;
        float v = 0.0f;
        if (row < NEXP)                 v = gate_W[row * DDIM + col];
        else if (row < NEXP + KLORA)    v = lora_down[(row - NEXP) * DDIM + col];
        Wg[idx] = v;
    } else {
        int j = idx - WGROWS * DDIM;
        if (j < DDIM * KLORA) {
            int o = j / KLORA;
            int k = j - o * KLORA;
            int e = k >> 2, r = k & 3;
            UpT[j] = lora_up[((size_t)e * DDIM + o) * RANK + r];
        }
    }
}

// ---------------------------------------------------------------------------
// Kernel 2: skinny WMMA GEMM  T[N x 64] = x[N x D] @ Wg^T
// Tile: 128 rows x 64 cols per workgroup; 8 waves, 16 rows each, 4 N-tiles.
// ---------------------------------------------------------------------------
__global__ void skinny_gemm_kernel(const float* __restrict__ x,
                                   const float* __restrict__ Wg,
                                   float* __restrict__ T) {
    __shared__ float As[TM * KT];        // [m][k]
    __shared__ float Bs[WGROWS * KT];    // [o][k]

    const int tid  = threadIdx.x;
    const int lane = tid & 31;
    const int w    = tid >> 5;           // 0..7
    const int half = lane >> 4;          // 0/1 -> K offset {0,2}
    const int l15  = lane & 15;
    const int nrow0 = blockIdx.x * TM;

    v8f acc[4] = {};

    for (int kc = 0; kc < NCHUNK; ++kc) {
        const int k0g = kc * KT;
        __syncthreads();
        for (int i = tid; i < TM * (KT / 4); i += 256) {       // 1024 float4
            int m = i >> 3, kq = i & 7;
            *(float4*)&As[m * KT + kq * 4] =
                *(const float4*)&x[(size_t)(nrow0 + m) * DDIM + k0g + kq * 4];
        }
        for (int i = tid; i < WGROWS * (KT / 4); i += 256) {   // 512 float4
            int o = i >> 3, kq = i & 7;
            *(float4*)&Bs[o * KT + kq * 4] =
                *(const float4*)&Wg[(size_t)o * DDIM + k0g + kq * 4];
        }
        __syncthreads();
#pragma unroll
        for (int k0 = 0; k0 < KT; k0 += 4) {
            v2f af = *(const v2f*)&As[(w * 16 + l15) * KT + k0 + half * 2];
#pragma unroll
            for (int b = 0; b < 4; ++b) {
                v2f bf = *(const v2f*)&Bs[(b * 16 + l15) * KT + k0 + half * 2];
                acc[b] = wmma_fp32(af, bf, acc[b]);
            }
        }
    }

#pragma unroll
    for (int b = 0; b < 4; ++b)
#pragma unroll
        for (int j = 0; j < 8; ++j)
            T[(size_t)(nrow0 + w * 16 + half * 8 + j) * WGROWS + b * 16 + l15] = acc[b][j];
}

// ---------------------------------------------------------------------------
// Kernel 3: top-2 gating softmax; fold weights*SCALING into Hw[N x 40]
// ---------------------------------------------------------------------------
__global__ void gate_kernel(const float* __restrict__ T,
                            const float* __restrict__ gate_b,
                            float* __restrict__ Hw) {
    const int n = blockIdx.x * 256 + threadIdx.x;
    float lg[NEXP];
#pragma unroll
    for (int e = 0; e < NEXP; ++e) lg[e] = T[(size_t)n * WGROWS + e] + gate_b[e];

    int i1 = 0;
#pragma unroll
    for (int e = 1; e < NEXP; ++e) if (lg[e] > lg[i1]) i1 = e;
    int i2 = (i1 == 0) ? 1 : 0;
#pragma unroll
    for (int e = 0; e < NEXP; ++e) if (e != i1 && lg[e] > lg[i2]) i2 = e;

    float e2  = expf(lg[i2] - lg[i1]);   // lg[i1] is the max
    float inv = 1.0f / (1.0f + e2);
    float w1 = inv, w2 = e2 * inv;

#pragma unroll
    for (int e = 0; e < NEXP; ++e) {
        float we = ((e == i1) ? w1 : (e == i2) ? w2 : 0.0f) * SCALING;
#pragma unroll
        for (int r = 0; r < RANK; ++r)
            Hw[(size_t)n * KLORA + e * 4 + r] =
                T[(size_t)n * WGROWS + NEXP + e * 4 + r] * we;
    }
}

// ---------------------------------------------------------------------------
// Kernel 4: main fused GEMM:
//   out[n][o] = b_base[o] + sum_d x[n][d]*W_base[o][d] + sum_k Hw[n][k]*UpT[o][k]
// 128x128 tile / workgroup; 8 wave32s in a 4x2 grid (each 32 rows x 64 cols).
// Double-buffered LDS fed by global_load_async_to_lds_b128 (ASYNCcnt pipelining).
// ---------------------------------------------------------------------------
#define BUFSZ (2 * TM * KT)    // floats per double-buffer slot pair member

__global__ void moe_lora_gemm_kernel(const float* __restrict__ x,
                                     const float* __restrict__ W_base,
                                     const float* __restrict__ b_base,
                                     const float* __restrict__ Hw,
                                     const float* __restrict__ UpT,
                                     float* __restrict__ out) {
    // 2 buffers x (As 128x32 + Bs 128x32) = 16384 floats = 64KB.
    // Rank-40 epilogue (Hs 128x40 + Us 128x40 = 40KB) overlays buffer 0/1 space.
    __shared__ float smem[2 * BUFSZ];

    const int tid  = threadIdx.x;
    const int lane = tid & 31;
    const int w    = tid >> 5;
    const int half = lane >> 4;
    const int l15  = lane & 15;
    const int wr   = w & 3;       // row group of 32
    const int wc   = w >> 2;      // col group of 64
    const int nrow0 = blockIdx.x * TM;
    const int ocol0 = blockIdx.y * TN;
    const unsigned lds_base = (unsigned)(uintptr_t)(void*)smem;

    v8f acc[2][4] = {};

    // ---- issue async copies for chunk kc into buffer (kc & 1) ----
    auto issue_chunk = [&](int kc) {
        const unsigned k0g   = (unsigned)kc * KT;
        const unsigned boff  = (unsigned)(kc & 1) * BUFSZ;   // floats
        for (int i = tid; i < TM * (KT / 4); i += 256) {     // 4 iters/thread
            int m = i >> 3, kq = i & 7;
            unsigned lds_a = lds_base + (boff + m * KT + kq * 4) * 4u;
            unsigned lds_b = lds_a + TM * KT * 4u;
            async_b128_to_lds(lds_a, x,
                ((unsigned)(nrow0 + m) * DDIM + k0g + kq * 4) * 4u);
            async_b128_to_lds(lds_b, W_base,
                ((unsigned)(ocol0 + m) * DDIM + k0g + kq * 4) * 4u);
        }
    };

    issue_chunk(0);
    for (int kc = 0; kc < NCHUNK; ++kc) {
        const float* As = smem + (kc & 1) * BUFSZ;
        const float* Bs = As + TM * KT;

        if (kc + 1 < NCHUNK) {
            issue_chunk(kc + 1);      // prefetch next chunk into other buffer
            WAIT_ASYNCCNT(8);         // 8 per chunk per wave; in-order completion
        } else {
            WAIT_ASYNCCNT(0);
        }
        __syncthreads();              // all waves' copies for chunk kc landed

#pragma unroll
        for (int k0 = 0; k0 < KT; k0 += 4) {
            v2f af[2], bf[4];
#pragma unroll
            for (int a = 0; a < 2; ++a)
                af[a] = *(const v2f*)&As[(wr * 32 + a * 16 + l15) * KT + k0 + half * 2];
#pragma unroll
            for (int b = 0; b < 4; ++b)
                bf[b] = *(const v2f*)&Bs[(wc * 64 + b * 16 + l15) * KT + k0 + half * 2];
#pragma unroll
            for (int a = 0; a < 2; ++a)
#pragma unroll
                for (int b = 0; b < 4; ++b)
                    acc[a][b] = wmma_fp32(af[a], bf[b], acc[a][b]);
        }
        __syncthreads();              // buffer may be overwritten next iteration
    }

    // ---- fused rank-40 LoRA-up epilogue (async-staged, overlays smem) ----
    float* Hs = smem;                 // [m][40]
    float* Us = smem + TM * KLORA;    // [o][40]
    for (int i = tid; i < TM * (KLORA / 4); i += 256) {      // 1280 float4
        int m = i / 10, kq = i - m * 10;
        unsigned lds_h = lds_base + (m * KLORA + kq * 4) * 4u;
        unsigned lds_u = lds_h + TM * KLORA * 4u;
        async_b128_to_lds(lds_h, Hw,
            ((unsigned)(nrow0 + m) * KLORA + kq * 4) * 4u);
        async_b128_to_lds(lds_u, UpT,
            ((unsigned)(ocol0 + m) * KLORA + kq * 4) * 4u);
    }
    WAIT_ASYNCCNT(0);
    __syncthreads();
#pragma unroll
    for (int k0 = 0; k0 < KLORA; k0 += 4) {
        v2f af[2], bf[4];
#pragma unroll
        for (int a = 0; a < 2; ++a)
            af[a] = *(const v2f*)&Hs[(wr * 32 + a * 16 + l15) * KLORA + k0 + half * 2];
#pragma unroll
        for (int b = 0; b < 4; ++b)
            bf[b] = *(const v2f*)&Us[(wc * 64 + b * 16 + l15) * KLORA + k0 + half * 2];
#pragma unroll
        for (int a = 0; a < 2; ++a)
#pragma unroll
            for (int b = 0; b < 4; ++b)
                acc[a][b] = wmma_fp32(af[a], bf[b], acc[a][b]);
    }

    // ---- store with bias ----
#pragma unroll
    for (int b = 0; b < 4; ++b) {
        const int col  = ocol0 + wc * 64 + b * 16 + l15;
        const float bias = b_base[col];
#pragma unroll
        for (int a = 0; a < 2; ++a) {
            const int r0 = nrow0 + wr * 32 + a * 16 + half * 8;
#pragma unroll
            for (int j = 0; j < 8; ++j)
                out[(size_t)(r0 + j) * DDIM + col] = acc[a][b][j] + bias;
        }
    }
}

// ---------------------------------------------------------------------------
extern "C" void kernel_launch(void* const* d_in, const int* in_sizes, int n_in,
                              void* d_out, int out_size, void* d_ws, size_t ws_size,
                              hipStream_t stream) {
    const float* x         = (const float*)d_in[0];
    const float* W_base    = (const float*)d_in[1];
    const float* b_base    = (const float*)d_in[2];
    const float* gate_W    = (const float*)d_in[3];
    const float* gate_b    = (const float*)d_in[4];
    const float* lora_down = (const float*)d_in[5];
    const float* lora_up   = (const float*)d_in[6];
    float* out = (float*)d_out;

    float* ws  = (float*)d_ws;
    float* Wg  = ws;                                   // 64*1024      = 65536
    float* UpT = Wg  + (size_t)WGROWS * DDIM;          // 1024*40      = 40960
    float* T   = UpT + (size_t)DDIM * KLORA;           // 16384*64     = 1048576
    float* Hw  = T   + (size_t)NTOK * WGROWS;          // 16384*40     = 655360
    // total ~7.3 MB of workspace

    pack_kernel<<<(WGROWS * DDIM + DDIM * KLORA) / 256, 256, 0, stream>>>(
        gate_W, lora_down, lora_up, Wg, UpT);
    skinny_gemm_kernel<<<NTOK / TM, 256, 0, stream>>>(x, Wg, T);
    gate_kernel<<<NTOK / 256, 256, 0, stream>>>(T, gate_b, Hw);
    moe_lora_gemm_kernel<<<dim3(NTOK / TM, DDIM / TN), 256, 0, stream>>>(
        x, W_base, b_base, Hw, UpT, out);
}